// edge_conv_69080253989612
// MI455X (gfx1250) — compile-verified
//
#include <hip/hip_runtime.h>

typedef __attribute__((ext_vector_type(2))) float v2f;
typedef __attribute__((ext_vector_type(8))) float v8f;

constexpr int B_   = 8;
constexpr int N_   = 4096;
constexpr int C_   = 64;
constexpr int O_   = 64;
constexpr int K_   = 20;
constexpr int CHUNK = 128;          // columns of the distance matrix per staging step
constexpr float BN_EPS = 1e-5f;
constexpr float SLOPE  = 0.2f;

// Async global->LDS copy, 16 bytes per lane (gfx1250, tracked by ASYNCcnt).
__device__ __forceinline__ void async_ld_b128(unsigned lds_off, const float* g) {
  unsigned long long ga = (unsigned long long)g;
  asm volatile("global_load_async_to_lds_b128 %0, %1, off"
               :: "v"(lds_off), "v"(ga) : "memory");
}
__device__ __forceinline__ void wait_async0() {
  asm volatile("s_wait_asynccnt 0" ::: "memory");
}
// Low 32 bits of a generic pointer to __shared__ = LDS byte offset.
__device__ __forceinline__ unsigned lds_off(const void* p) {
  return (unsigned)(size_t)p;
}

// ---------------------------------------------------------------------------
// Kernel 1: per-point squared norm
// ---------------------------------------------------------------------------
__global__ __launch_bounds__(256) void sq_kernel(const float* __restrict__ P,
                                                 float* __restrict__ sq) {
  int t = blockIdx.x * 256 + threadIdx.x;           // 0 .. B*N-1
  const float4* p4 = (const float4*)(P + (size_t)t * C_);
  float s = 0.f;
#pragma unroll
  for (int i = 0; i < C_ / 4; ++i) {
    float4 v = p4[i];
    s += v.x * v.x + v.y * v.y + v.z * v.z + v.w * v.w;
  }
  sq[t] = s;
}

// ---------------------------------------------------------------------------
// Kernel 2: fused distance-GEMM (WMMA f32 16x16x4) + dilated top-k.
// One block per (batch, 16-row tile). 8 waves each own a 16x16 column tile.
// Chunk points are double-buffered and fetched with async global->LDS loads;
// selection uses a wave32 ballot pre-filter (2 rows per wave).
// ---------------------------------------------------------------------------
__global__ __launch_bounds__(256) void knn_kernel(const float* __restrict__ P,
                                                  const float* __restrict__ sq,
                                                  int* __restrict__ nn_idx) {
  __shared__ float lsA[16][68];             // A rows (16 x 64), padded
  __shared__ float lsSqA[16];
  __shared__ float lsP[2][CHUNK][68];       // double-buffered chunk points
  __shared__ float lsSqB[2][CHUNK];
  __shared__ float lsD[16][132];            // distance tile 16 x 128, padded
  __shared__ float topd[16][40];
  __shared__ int   topi[16][40];

  const int tid  = threadIdx.x;
  const int wave = tid >> 5;
  const int lane = tid & 31;
  const int b    = blockIdx.x >> 8;                 // N/16 = 256 tiles per batch
  const int row0 = (blockIdx.x & 255) << 4;
  const float* Pb  = P  + (size_t)b * N_ * C_;
  const float* sqb = sq + (size_t)b * N_;

  // stage A tile + init top-k
  for (int i = tid; i < 16 * C_; i += 256) {
    int r = i >> 6, c = i & 63;
    lsA[r][c] = Pb[(size_t)(row0 + r) * C_ + c];
  }
  if (tid < 16) lsSqA[tid] = sqb[row0 + tid];
  for (int i = tid; i < 16 * 40; i += 256) {
    topd[i / 40][i % 40] = 3.0e38f;
    topi[i / 40][i % 40] = 0;
  }

  // prologue: async-stage chunk 0 into buffer 0
  {
    const float* g0 = Pb;                            // m0 = 0
#pragma unroll
    for (int i = 0; i < 8; ++i) {                    // 2048 float4s / 256 thr
      int f  = tid + i * 256;
      int r  = f >> 4, c4 = f & 15;
      async_ld_b128(lds_off(&lsP[0][r][c4 * 4]), g0 + (size_t)r * C_ + c4 * 4);
    }
    if (tid < CHUNK) lsSqB[0][tid] = sqb[tid];
  }
  wait_async0();
  __syncthreads();

  const int mlo  = lane & 15;
  const int khi  = (lane & 16) ? 2 : 0;
  const int moff = (lane & 16) ? 8 : 0;
  const int col0 = wave << 4;

  for (int m0 = 0; m0 < N_; m0 += CHUNK) {
    const int cur = (m0 / CHUNK) & 1;

    // issue async loads for the NEXT chunk into the other buffer
    if (m0 + CHUNK < N_) {
      const float* gn = Pb + (size_t)(m0 + CHUNK) * C_;
#pragma unroll
      for (int i = 0; i < 8; ++i) {
        int f  = tid + i * 256;
        int r  = f >> 4, c4 = f & 15;
        async_ld_b128(lds_off(&lsP[1 - cur][r][c4 * 4]),
                      gn + (size_t)r * C_ + c4 * 4);
      }
      if (tid < CHUNK) lsSqB[1 - cur][tid] = sqb[m0 + CHUNK + tid];
    }

    // 16x16 inner products: two independent 8-step WMMA chains (K=64)
    v8f acc0 = {}, acc1 = {};
#pragma unroll
    for (int ks = 0; ks < 8; ++ks) {
      int kc0 = ks * 4 + khi;
      int kc1 = kc0 + 32;
      v2f a0, b0, a1, b1;
      a0.x = lsA[mlo][kc0];             a0.y = lsA[mlo][kc0 + 1];
      b0.x = lsP[cur][col0 + mlo][kc0]; b0.y = lsP[cur][col0 + mlo][kc0 + 1];
      a1.x = lsA[mlo][kc1];             a1.y = lsA[mlo][kc1 + 1];
      b1.x = lsP[cur][col0 + mlo][kc1]; b1.y = lsP[cur][col0 + mlo][kc1 + 1];
      acc0 = __builtin_amdgcn_wmma_f32_16x16x4_f32(false, a0, false, b0,
                                                   (short)0, acc0, false, false);
      acc1 = __builtin_amdgcn_wmma_f32_16x16x4_f32(false, a1, false, b1,
                                                   (short)0, acc1, false, false);
    }

    // inner -> squared distance, into LDS
#pragma unroll
    for (int v = 0; v < 8; ++v) {
      int M   = v + moff;
      int col = col0 + mlo;
      lsD[M][col] = lsSqA[M] - 2.0f * (acc0[v] + acc1[v]) + lsSqB[cur][col];
    }
    __syncthreads();

    // top-40 selection: each wave owns 2 rows; ballot pre-filter, rare
    // insertions serialized on lane 0 (ascending order => stable ties).
#pragma unroll 1
    for (int rr = 0; rr < 2; ++rr) {
      int r = wave * 2 + rr;
#pragma unroll 1
      for (int i0 = 0; i0 < CHUNK; i0 += 32) {
        float dv  = lsD[r][i0 + lane];
        float thr = topd[r][39];
        unsigned mask = (unsigned)__ballot(dv < thr);
        while (mask) {
          int i = __ffs(mask) - 1;
          mask &= mask - 1;
          if (lane == 0) {
            float v = lsD[r][i0 + i];
            if (v < topd[r][39]) {
              int j = 39;
              while (j > 0 && topd[r][j - 1] > v) {
                topd[r][j] = topd[r][j - 1];
                topi[r][j] = topi[r][j - 1];
                --j;
              }
              topd[r][j] = v;
              topi[r][j] = m0 + i0 + i;
            }
          }
        }
      }
    }

    // this wave's async loads for the next chunk must have landed
    wait_async0();
    __syncthreads();
  }

  // dilation d=2: keep positions 0,2,...,38
  if (tid < 16 * K_) {
    int r = tid / K_, j = tid - r * K_;
    nn_idx[((size_t)b * N_ + row0 + r) * K_ + j] = topi[r][j * 2];
  }
}

// ---------------------------------------------------------------------------
// Kernel 3: gather edge features + (R x 64) = (R x 128) @ Wt(128 x 64) WMMA
// GEMM + fused bias/BN/LeakyReLU. One block = 128 rows (8 waves x 16 rows).
// ---------------------------------------------------------------------------
__global__ __launch_bounds__(256) void edgeconv_kernel(
    const float* __restrict__ P, const int* __restrict__ nn_idx,
    const float* __restrict__ Wg, const float* __restrict__ cb,
    const float* __restrict__ gamma, const float* __restrict__ beta,
    const float* __restrict__ mean, const float* __restrict__ var,
    float* __restrict__ out) {
  __shared__ float lsW[2 * C_][65];     // W^T : [k=128][o=64], padded
  __shared__ float lsE[128][130];       // edge rows for whole block, padded

  const int tid  = threadIdx.x;
  const int wave = tid >> 5;
  const int lane = tid & 31;
  const int b        = blockIdx.x / 640;           // 81920/128 = 640 blocks/batch
  const int rowBlock = (blockIdx.x % 640) * 128;
  const float* Pb = P + (size_t)b * N_ * C_;

  // stage W^T (coalesced global reads, padded LDS writes)
  for (int i = tid; i < O_ * 2 * C_; i += 256) {
    int o = i >> 7, c = i & 127;
    lsW[c][o] = Wg[i];                  // Wg is [o][c] row-major
  }

  // stage this wave's 16 edge rows: [central(64) | nbr-central(64)]
  const int rbase = rowBlock + wave * 16;
#pragma unroll 1
  for (int r = 0; r < 16; ++r) {
    int rowG = rbase + r;
    int n  = rowG / K_;
    int kk = rowG - n * K_;
    int jn = nn_idx[((size_t)b * N_ + n) * K_ + kk];
    const float* cen = Pb + (size_t)n  * C_;
    const float* nbr = Pb + (size_t)jn * C_;
#pragma unroll
    for (int c = lane; c < C_; c += 32) {
      float cv = cen[c];
      lsE[wave * 16 + r][c]      = cv;
      lsE[wave * 16 + r][C_ + c] = nbr[c] - cv;
    }
  }
  __syncthreads();

  const int mlo  = lane & 15;
  const int khi  = (lane & 16) ? 2 : 0;
  const int moff = (lane & 16) ? 8 : 0;

  v8f acc[4] = {v8f{}, v8f{}, v8f{}, v8f{}};
#pragma unroll
  for (int ks = 0; ks < (2 * C_) / 4; ++ks) {       // 32 K-steps
    int kc = ks * 4 + khi;
    v2f a;
    a.x = lsE[wave * 16 + mlo][kc];
    a.y = lsE[wave * 16 + mlo][kc + 1];
#pragma unroll
    for (int t = 0; t < 4; ++t) {
      v2f bf;
      bf.x = lsW[kc][t * 16 + mlo];
      bf.y = lsW[kc + 1][t * 16 + mlo];
      acc[t] = __builtin_amdgcn_wmma_f32_16x16x4_f32(false, a, false, bf,
                                                     (short)0, acc[t], false, false);
    }
  }

  // fused bias + BN + LeakyReLU, coalesced stores
#pragma unroll
  for (int t = 0; t < 4; ++t) {
    int o = t * 16 + mlo;
    float inv   = gamma[o] / sqrtf(var[o] + BN_EPS);
    float shift = cb[o] - mean[o];
    float bt    = beta[o];
#pragma unroll
    for (int v = 0; v < 8; ++v) {
      int M = v + moff;
      float y = (acc[t][v] + shift) * inv + bt;
      y = y > 0.f ? y : SLOPE * y;
      out[((size_t)b * (N_ * K_) + (rbase + M)) * O_ + o] = y;
    }
  }
}

// ---------------------------------------------------------------------------
extern "C" void kernel_launch(void* const* d_in, const int* in_sizes, int n_in,
                              void* d_out, int out_size, void* d_ws, size_t ws_size,
                              hipStream_t stream) {
  const float* P     = (const float*)d_in[0];
  const float* Wg    = (const float*)d_in[1];
  const float* cb    = (const float*)d_in[2];
  const float* gamma = (const float*)d_in[3];
  const float* beta  = (const float*)d_in[4];
  const float* mean  = (const float*)d_in[5];
  const float* var   = (const float*)d_in[6];
  float* out = (float*)d_out;

  float* sq = (float*)d_ws;                                   // B*N floats
  int*   nn = (int*)((char*)d_ws + sizeof(float) * B_ * N_);  // B*N*K ints

  sq_kernel<<<(B_ * N_) / 256, 256, 0, stream>>>(P, sq);
  knn_kernel<<<B_ * (N_ / 16), 256, 0, stream>>>(P, sq, nn);
  edgeconv_kernel<<<B_ * 640, 256, 0, stream>>>(P, nn, Wg, cb, gamma, beta,
                                                mean, var, out);
}